// LieSE2_66649302499847
// MI455X (gfx1250) — compile-verified
//
#include <hip/hip_runtime.h>
#include <math.h>

typedef __attribute__((ext_vector_type(4))) float v4f;
typedef int v4i __attribute__((vector_size(16)));

#define AS1 __attribute__((address_space(1)))
#define AS3 __attribute__((address_space(3)))

#ifndef __has_builtin
#define __has_builtin(x) 0
#endif

// Async global->LDS path only in device compilation and only if the gfx1250
// builtins exist in this toolchain; otherwise fall back to direct NT loads.
#if defined(__HIP_DEVICE_COMPILE__) && __has_builtin(__builtin_amdgcn_global_load_async_to_lds_b128)
#define HAVE_ASYNC 1
#else
#define HAVE_ASYNC 0
#endif

#if HAVE_ASYNC
#if __has_builtin(__builtin_amdgcn_s_wait_asynccnt)
#define WAIT_ASYNC(N)                                                          \
  do {                                                                         \
    __builtin_amdgcn_s_wait_asynccnt(N);                                       \
    asm volatile("" ::: "memory");                                             \
  } while (0)
#else
#define WAIT_ASYNC(N) asm volatile("s_wait_asynccnt %0" ::"n"(N) : "memory")
#endif
#define ASYNC_LD_B128(gp, lp)                                                  \
  __builtin_amdgcn_global_load_async_to_lds_b128((AS1 v4i*)(gp),               \
                                                 (AS3 v4i*)(lp), 0, 0)
#endif

#define EPS_F 1e-5f
#define BLOCK 256
#define EPT 4
#define TILE (BLOCK * EPT)     /* 1024 elements per tile           */
#define TILE_BYTES (TILE * 12) /* 12288 B of input per tile        */
#define GRID_BLOCKS 2048

// One SE(2) exp-map element -> 9 output floats (row-major 3x3).
__device__ __forceinline__ void compute9(float ux, float uy, float wz,
                                         float* __restrict__ o) {
  float th2 = wz * wz;
  float th = sqrtf(th2);
  float s = sinf(th);
  float c = cosf(th);
  float A = s / (th + EPS_F);
  float rt = 1.0f / (th2 + EPS_F);
  float Bc = (1.0f - c) * rt;
  float Cc = (1.0f - A) * rt;
  float r00 = 1.0f - Bc * th2;
  float Aw = A * wz;
  float v00 = 1.0f - Cc * th2;
  float Bw = Bc * wz;
  float tx = v00 * ux - Bw * uy;
  float ty = Bw * ux + v00 * uy;
  o[0] = r00; o[1] = -Aw; o[2] = tx;
  o[3] = Aw;  o[4] = r00; o[5] = ty;
  o[6] = 0.f; o[7] = 0.f; o[8] = 1.f;
}

// Consume 4 packed elements (12 floats in three v4f) and emit 36 floats
// (144 contiguous bytes) via nine 128-bit non-temporal stores.
__device__ __forceinline__ void consume_store(v4f f0, v4f f1, v4f f2,
                                              float* __restrict__ out,
                                              size_t e0) {
  float u[4][3] = {
      {f0.x, f0.y, f0.z},
      {f0.w, f1.x, f1.y},
      {f1.z, f1.w, f2.x},
      {f2.y, f2.z, f2.w},
  };
  float r[4][9];
#pragma unroll
  for (int j = 0; j < 4; ++j) compute9(u[j][0], u[j][1], u[j][2], r[j]);
  const float* rf = &r[0][0];
  v4f* dst = (v4f*)(out + 9 * e0);
#pragma unroll
  for (int k = 0; k < 9; ++k) {
    v4f q = {rf[4 * k + 0], rf[4 * k + 1], rf[4 * k + 2], rf[4 * k + 3]};
    __builtin_nontemporal_store(q, dst + k);
  }
}

__global__ __launch_bounds__(BLOCK) void se2_exp_kernel(
    const float* __restrict__ uv, float* __restrict__ out, int n) {
  const int tid = (int)threadIdx.x;
  const int numTiles = n / TILE;

#if HAVE_ASYNC
  // Double-buffered async global->LDS pipeline. Each lane copies exactly the
  // 48 B it will consume, so no cross-wave sharing -> no workgroup barrier;
  // ASYNCcnt ordering alone is sufficient (async loads complete in order).
  __shared__ __align__(16) char smem[2][TILE_BYTES];
  int tile = (int)blockIdx.x;
  int buf = 0;
  if (tile < numTiles) {
    char* g = (char*)uv + (size_t)tile * TILE_BYTES + (size_t)tid * 48u;
    char* l = &smem[0][(size_t)tid * 48u];
#pragma unroll
    for (int k = 0; k < 3; ++k) ASYNC_LD_B128(g + 16 * k, l + 16 * k);
  }
  for (; tile < numTiles; tile += (int)gridDim.x) {
    const int nxt = tile + (int)gridDim.x;
    if (nxt < numTiles) {
      char* g = (char*)uv + (size_t)nxt * TILE_BYTES + (size_t)tid * 48u;
      char* l = &smem[buf ^ 1][(size_t)tid * 48u];
#pragma unroll
      for (int k = 0; k < 3; ++k) ASYNC_LD_B128(g + 16 * k, l + 16 * k);
      WAIT_ASYNC(3); // 3 newest may be in flight; current tile's 3 are done
    } else {
      WAIT_ASYNC(0);
    }
    const v4f* src = (const v4f*)&smem[buf][(size_t)tid * 48u];
    v4f f0 = src[0], f1 = src[1], f2 = src[2];
    consume_store(f0, f1, f2, out,
                  (size_t)tile * TILE + (size_t)tid * EPT);
    buf ^= 1;
  }
#else
  // Fallback: direct 128-bit non-temporal streaming loads + prefetch.
  for (int tile = (int)blockIdx.x; tile < numTiles; tile += (int)gridDim.x) {
    const char* g = (const char*)uv + (size_t)tile * TILE_BYTES + (size_t)tid * 48u;
    const int nxt = tile + (int)gridDim.x;
    if (nxt < numTiles)
      __builtin_prefetch(
          (const char*)uv + (size_t)nxt * TILE_BYTES + (size_t)tid * 48u, 0, 0);
    v4f f0 = __builtin_nontemporal_load((const v4f*)g);
    v4f f1 = __builtin_nontemporal_load((const v4f*)(g + 16));
    v4f f2 = __builtin_nontemporal_load((const v4f*)(g + 32));
    consume_store(f0, f1, f2, out,
                  (size_t)tile * TILE + (size_t)tid * EPT);
  }
#endif

  // Scalar tail (n is a multiple of TILE for this problem, but stay generic).
  for (int i = numTiles * TILE + (int)(blockIdx.x * BLOCK) + tid; i < n;
       i += (int)(gridDim.x * BLOCK)) {
    float o[9];
    compute9(uv[3 * (size_t)i], uv[3 * (size_t)i + 1], uv[3 * (size_t)i + 2], o);
#pragma unroll
    for (int k = 0; k < 9; ++k) out[9 * (size_t)i + k] = o[k];
  }
}

extern "C" void kernel_launch(void* const* d_in, const int* in_sizes, int n_in,
                              void* d_out, int out_size, void* d_ws,
                              size_t ws_size, hipStream_t stream) {
  (void)n_in; (void)out_size; (void)d_ws; (void)ws_size;
  const float* uv = (const float*)d_in[0];
  float* out = (float*)d_out;
  const int n = in_sizes[0] / 3;
  se2_exp_kernel<<<dim3(GRID_BLOCKS), dim3(BLOCK), 0, stream>>>(uv, out, n);
}